// TopKRouter_90933047591232
// MI455X (gfx1250) — compile-verified
//
#include <hip/hip_runtime.h>

typedef __attribute__((ext_vector_type(2))) float v2f;
typedef __attribute__((ext_vector_type(4))) float v4f;
typedef __attribute__((ext_vector_type(8))) float v8f;

#define DIM   4096
#define NEXP  64
#define TOPK  8
#define BM    32     // tokens per block
#define KC    128    // K chunk
#define LDP   132    // padded LDS row stride (floats) -> conflict-free ds_load_b64
#define NC    (DIM / KC)

// -------------------- GEMM: scores = H (T x D) * W^T (D x E) --------------------
// Double-buffered LDS pipeline: next chunk's HBM loads are issued before the
// current chunk's WMMA loop so the 512 MB hidden_states stream stays in flight.
__global__ __launch_bounds__(256) void router_gemm(const float* __restrict__ H,
                                                   const float* __restrict__ W,
                                                   float* __restrict__ S, int T) {
  __shared__ float As[2][BM * LDP];    // 2 x 16896 B
  __shared__ float Bs[2][NEXP * LDP];  // 2 x 33792 B
  const int tid   = threadIdx.x;
  const int lane  = tid & 31;
  const int wave  = tid >> 5;
  const int mtile = wave >> 2;   // 0..1  (16-token tile)
  const int etile = wave & 3;    // 0..3  (16-expert tile)
  const int half  = lane >> 4;   // 0..1
  const int mrow  = lane & 15;
  const int block_m = blockIdx.x * BM;

  v8f acc0 = {}, acc1 = {};
  v4f ra[4], rb[8];

  // ---- register staging: A = 32x128 (4 x b128/thread), B = 64x128 (8 x b128/thread)
#define LOAD_REGS(KCOFF)                                                        \
  do {                                                                          \
    _Pragma("unroll")                                                           \
    for (int j = 0; j < 4; ++j) {                                               \
      int v = tid + 256 * j; int row = v >> 5; int c4 = v & 31;                 \
      ra[j] = __builtin_nontemporal_load(                                       \
          (const v4f*)(H + (size_t)(block_m + row) * DIM + (KCOFF) + c4 * 4));  \
    }                                                                           \
    _Pragma("unroll")                                                           \
    for (int j = 0; j < 8; ++j) {                                               \
      int v = tid + 256 * j; int row = v >> 5; int c4 = v & 31;                 \
      rb[j] = *(const v4f*)(W + (size_t)row * DIM + (KCOFF) + c4 * 4);          \
    }                                                                           \
  } while (0)

#define STORE_REGS(BUF)                                                         \
  do {                                                                          \
    _Pragma("unroll")                                                           \
    for (int j = 0; j < 4; ++j) {                                               \
      int v = tid + 256 * j; int row = v >> 5; int c4 = v & 31;                 \
      *(v4f*)(&As[BUF][row * LDP + c4 * 4]) = ra[j];                            \
    }                                                                           \
    _Pragma("unroll")                                                           \
    for (int j = 0; j < 8; ++j) {                                               \
      int v = tid + 256 * j; int row = v >> 5; int c4 = v & 31;                 \
      *(v4f*)(&Bs[BUF][row * LDP + c4 * 4]) = rb[j];                            \
    }                                                                           \
  } while (0)

  // prologue: stage chunk 0
  LOAD_REGS(0);
  STORE_REGS(0);
  __syncthreads();

  for (int c = 0; c < NC; ++c) {
    const int cur = c & 1;
    if (c + 1 < NC) LOAD_REGS((c + 1) * KC);   // HBM loads in flight during WMMA

    const float* Ab = &As[cur][(mtile * 16 + mrow) * LDP + 2 * half];
    const float* Bb = &Bs[cur][(etile * 16 + mrow) * LDP + 2 * half];
#pragma unroll
    for (int k = 0; k < KC; k += 8) {
      // A layout: v0 = A[m][k + 2*half], v1 = A[m][k+1 + 2*half]
      // B layout: v0 = B[k + 2*half][n] = W[n][k + 2*half]
      v2f a0 = *(const v2f*)(Ab + k);
      v2f b0 = *(const v2f*)(Bb + k);
      acc0 = __builtin_amdgcn_wmma_f32_16x16x4_f32(false, a0, false, b0,
                                                   (short)0, acc0, false, false);
      v2f a1 = *(const v2f*)(Ab + k + 4);
      v2f b1 = *(const v2f*)(Bb + k + 4);
      acc1 = __builtin_amdgcn_wmma_f32_16x16x4_f32(false, a1, false, b1,
                                                   (short)0, acc1, false, false);
    }

    if (c + 1 < NC) {
      STORE_REGS((c + 1) & 1);
      __syncthreads();
    }
  }
#undef LOAD_REGS
#undef STORE_REGS

  // C/D layout: VGPR r -> M = 8*half + r, N = lane&15
  const int n = lane & 15;
#pragma unroll
  for (int r = 0; r < 8; ++r) {
    int m = mtile * 16 + half * 8 + r;
    S[(size_t)(block_m + m) * NEXP + etile * 16 + n] = acc0[r] + acc1[r];
  }
}

// -------------------- softmax + biased top-8 + renorm, one wave32 per token --------------------
__global__ __launch_bounds__(256) void router_topk(const float* __restrict__ S,
                                                   const float* __restrict__ bias,
                                                   int* __restrict__ oidx,
                                                   float* __restrict__ oprob, int T) {
  const int lane = threadIdx.x & 31;
  const int wave = threadIdx.x >> 5;
  const int t = blockIdx.x * 8 + wave;
  if (t >= T) return;

  v2f s  = *(const v2f*)(S + (size_t)t * NEXP + lane * 2);   // experts 2*lane, 2*lane+1
  v2f bb = *(const v2f*)(bias + lane * 2);

  // softmax over 64 experts (wave32, 2 per lane)
  float m = fmaxf(s.x, s.y);
#pragma unroll
  for (int o = 16; o >= 1; o >>= 1) m = fmaxf(m, __shfl_xor(m, o, 32));
  float e0 = __expf(s.x - m), e1 = __expf(s.y - m);
  float sum = e0 + e1;
#pragma unroll
  for (int o = 16; o >= 1; o >>= 1) sum += __shfl_xor(sum, o, 32);
  float inv = 1.0f / sum;
  float p0 = e0 * inv, p1 = e1 * inv;
  float k0 = p0 + bb.x, k1 = p1 + bb.y;   // biased selection keys

  int   sel_i[TOPK];
  float sel_p[TOPK];
  float psum = 0.f;
#pragma unroll
  for (int j = 0; j < TOPK; ++j) {
    float v; int i;
    if (k0 >= k1) { v = k0; i = lane * 2; } else { v = k1; i = lane * 2 + 1; }
#pragma unroll
    for (int o = 16; o >= 1; o >>= 1) {
      float ov = __shfl_xor(v, o, 32);
      int   oi = __shfl_xor(i, o, 32);
      if (ov > v || (ov == v && oi < i)) { v = ov; i = oi; }   // ties -> lowest index
    }
    float cand = (i & 1) ? p1 : p0;          // i is wave-uniform after reduction
    float pv = __shfl(cand, i >> 1, 32);     // unbiased prob of winner
    sel_i[j] = i; sel_p[j] = pv; psum += pv;
    if (lane == (i >> 1)) { if (i & 1) k1 = -3.4e38f; else k0 = -3.4e38f; }
  }
  float rinv = 1.0f / (psum + 1e-20f);
  if (lane == 0) {
#pragma unroll
    for (int j = 0; j < TOPK; ++j) {
      oidx [(size_t)t * TOPK + j] = sel_i[j];
      oprob[(size_t)t * TOPK + j] = sel_p[j] * rinv;
    }
  }
}

extern "C" void kernel_launch(void* const* d_in, const int* in_sizes, int n_in,
                              void* d_out, int out_size, void* d_ws, size_t ws_size,
                              hipStream_t stream) {
  const float* H = (const float*)d_in[0];   // (T, 4096)
  const float* W = (const float*)d_in[1];   // (64, 4096)
  const float* B = (const float*)d_in[2];   // (64,)
  const int T = in_sizes[0] / DIM;          // 32768

  float* scores = (float*)d_ws;             // T*64 f32 = 8 MB scratch
  const int halfOut = out_size / 2;         // T*8
  int*   oidx  = (int*)d_out;               // first: int32 indices
  float* oprob = (float*)d_out + halfOut;   // second: f32 probs

  router_gemm<<<T / BM, 256, 0, stream>>>(H, W, scores, T);
  router_topk<<<(T + 7) / 8, 256, 0, stream>>>(scores, B, oidx, oprob, T);
}